// BidirectionalSoftmax_10763188043926
// MI455X (gfx1250) — compile-verified
//
#include <hip/hip_runtime.h>
#include <math.h>
#include <stdint.h>

typedef float v4f __attribute__((ext_vector_type(4)));
typedef int   v4i __attribute__((vector_size(16)));   // matches builtin's V4i

#define TAU_INV 2.0f      // 1/0.5
#define EPS_V   1e-8f
#define NEG_F  -1e30f

constexpr int L1 = 2048;
constexpr int L2 = 2048;

// ---- gfx1250 async global->LDS path (guarded; fallback = NT loads) ----
#if defined(__AMDGCN__) && defined(__gfx1250__)
# if __has_builtin(__builtin_amdgcn_global_load_async_to_lds_b128) && \
     __has_builtin(__builtin_amdgcn_s_wait_asynccnt)
#  define HAVE_ASYNC_LDS 1
# endif
#endif
#ifndef HAVE_ASYNC_LDS
# define HAVE_ASYNC_LDS 0
#endif

#if HAVE_ASYNC_LDS
__device__ __forceinline__ void async_copy16(v4f* lds_dst, const float* g_src) {
  __builtin_amdgcn_global_load_async_to_lds_b128(
      (__attribute__((address_space(1))) v4i*)g_src,
      (__attribute__((address_space(3))) v4i*)lds_dst,
      0, 0);
}
#endif

// Online (max, rescaled sum-of-exp) update; inputs are raw sim values
// (scaled by 1/TAU inside). Finite -1e30 sentinel => exp() underflows to
// exactly 0, so no NaN guards are ever needed.
__device__ __forceinline__ void online4(float& m, float& s,
                                        float a, float b, float c, float d) {
  float x0 = a * TAU_INV, x1 = b * TAU_INV, x2 = c * TAU_INV, x3 = d * TAU_INV;
  float mx = fmaxf(fmaxf(x0, x1), fmaxf(x2, x3));
  float mn = fmaxf(m, mx);
  s = s * __expf(m - mn) + (__expf(x0 - mn) + __expf(x1 - mn))
                         + (__expf(x2 - mn) + __expf(x3 - mn));
  m = mn;
}
__device__ __forceinline__ void online1(float& m, float& s, float a) {
  float x = a * TAU_INV;
  float mn = fmaxf(m, x);
  s = s * __expf(m - mn) + __expf(x - mn);
  m = mn;
}

// ---------------- Kernel 1: per-row (max, sumexp); one wave32 per row ----
__global__ __launch_bounds__(256) void row_stats_kernel(
    const float* __restrict__ sim, const int* __restrict__ lengths,
    float* __restrict__ rowM, float* __restrict__ rowS)
{
  const int lane = threadIdx.x & 31;
  const int row  = blockIdx.x * 8 + (threadIdx.x >> 5);   // 8 waves / block
  const int b    = row >> 11;                              // /L1
  const int l1v  = lengths[2 * b + 1];
  const v4f* p = reinterpret_cast<const v4f*>(sim + (size_t)row * L2);

  float m = NEG_F, s = 0.0f;
  const int cmax = (l1v + 127) >> 7;   // 128-column chunks; skip fully-masked tail
  #pragma unroll 4
  for (int c = 0; c < cmax; ++c) {
    v4f v = p[c * 32 + lane];
    const int j0 = c * 128 + lane * 4;
    float a0 = (j0 + 0 < l1v) ? v.x : NEG_F;
    float a1 = (j0 + 1 < l1v) ? v.y : NEG_F;
    float a2 = (j0 + 2 < l1v) ? v.z : NEG_F;
    float a3 = (j0 + 3 < l1v) ? v.w : NEG_F;
    online4(m, s, a0, a1, a2, a3);
  }
  // wave32 butterfly combine
  #pragma unroll
  for (int off = 16; off; off >>= 1) {
    float mo = __shfl_xor(m, off, 32);
    float so = __shfl_xor(s, off, 32);
    float mn = fmaxf(m, mo);
    s = s * __expf(m - mn) + so * __expf(mo - mn);
    m = mn;
  }
  if (lane == 0) { rowM[row] = m; rowS[row] = s; }
}

// ---- Kernel 2: per-column partial stats; thread owns 4 cols, i-split ----
__global__ __launch_bounds__(256) void col_stats_partial_kernel(
    const float* __restrict__ sim, const int* __restrict__ lengths,
    float* __restrict__ partM, float* __restrict__ partS,
    int rps, int ncols)
{
  const int g  = blockIdx.x * 1024 + threadIdx.x * 4;  // global column id
  const int sp = blockIdx.y;
  const int b  = g >> 11;
  const int j  = g & (L2 - 1);
  const int l0 = lengths[2 * b];

  int i   = sp * rps;
  int end = i + rps; if (end > l0) end = l0;

  float m0 = NEG_F, m1 = NEG_F, m2 = NEG_F, m3 = NEG_F;
  float s0 = 0.f,  s1 = 0.f,  s2 = 0.f,  s3 = 0.f;
  const float* base = sim + (size_t)b * L1 * L2 + j;

  for (; i + 4 <= end; i += 4) {
    if (i + 32 < end)
      __builtin_prefetch(base + (size_t)(i + 32) * L2, 0, 3); // global_prefetch_b8
    v4f r0 = *(const v4f*)(base + (size_t)(i + 0) * L2);
    v4f r1 = *(const v4f*)(base + (size_t)(i + 1) * L2);
    v4f r2 = *(const v4f*)(base + (size_t)(i + 2) * L2);
    v4f r3 = *(const v4f*)(base + (size_t)(i + 3) * L2);
    online4(m0, s0, r0.x, r1.x, r2.x, r3.x);
    online4(m1, s1, r0.y, r1.y, r2.y, r3.y);
    online4(m2, s2, r0.z, r1.z, r2.z, r3.z);
    online4(m3, s3, r0.w, r1.w, r2.w, r3.w);
  }
  for (; i < end; ++i) {
    v4f r = *(const v4f*)(base + (size_t)i * L2);
    online1(m0, s0, r.x); online1(m1, s1, r.y);
    online1(m2, s2, r.z); online1(m3, s3, r.w);
  }
  const size_t o = (size_t)sp * ncols + g;
  v4f mv = {m0, m1, m2, m3};
  v4f sv = {s0, s1, s2, s3};
  *(v4f*)(partM + o) = mv;
  *(v4f*)(partS + o) = sv;
}

// ---------------- Kernel 3: combine the i-split partials -----------------
__global__ __launch_bounds__(256) void col_combine_kernel(
    const float* __restrict__ partM, const float* __restrict__ partS,
    float* __restrict__ colM, float* __restrict__ colS, int split, int ncols)
{
  const int g = blockIdx.x * blockDim.x + threadIdx.x;
  if (g >= ncols) return;
  float m = NEG_F, s = 0.f;
  for (int sp = 0; sp < split; ++sp) {
    float mo = partM[(size_t)sp * ncols + g];
    float so = partS[(size_t)sp * ncols + g];
    float mn = fmaxf(m, mo);
    s = s * __expf(m - mn) + so * __expf(mo - mn);
    m = mn;
  }
  colM[g] = m; colS[g] = s;
}

// ---- Kernel 4: fused output. Block = 16 rows x 1024 cols; col stats in
// registers; sim rows staged via async global->LDS double buffer if present.
__global__ __launch_bounds__(256) void bidir_out_kernel(
    const float* __restrict__ sim, const int* __restrict__ lengths,
    const float* __restrict__ rowM, const float* __restrict__ rowS,
    const float* __restrict__ colM, const float* __restrict__ colS,
    float* __restrict__ out)
{
  constexpr int RI = 16;
  constexpr int CT = L2 / 1024;   // 2 column tiles
  constexpr int RT = L1 / RI;     // 128 row tiles
  const int bid = blockIdx.x;
  const int ct  = bid % CT;
  const int rt  = (bid / CT) % RT;
  const int b   = bid / (CT * RT);
  const int tid = threadIdx.x;
  const int j   = ct * 1024 + tid * 4;
  const int i0  = rt * RI;
  const int l0  = lengths[2 * b];
  const int l1v = lengths[2 * b + 1];

  float* orow = out + ((size_t)(b * L1 + i0) * L2 + j);
  const v4f z = {0.f, 0.f, 0.f, 0.f};

  if (i0 >= l0) {   // uniform per block: whole tile masked out
    #pragma unroll
    for (int r = 0; r < RI; ++r) {
      __builtin_nontemporal_store(z, (v4f*)orow);
      orow += L2;
    }
    return;
  }

  const int gc = b * L2 + j;
  v4f cm = *(const v4f*)(colM + gc);
  v4f cs = *(const v4f*)(colS + gc);
  float rc0 = 1.0f / cs.x, rc1 = 1.0f / cs.y, rc2 = 1.0f / cs.z, rc3 = 1.0f / cs.w;
  const bool vj0 = (j + 0) < l1v, vj1 = (j + 1) < l1v,
             vj2 = (j + 2) < l1v, vj3 = (j + 3) < l1v;

  const float* srow = sim + ((size_t)(b * L1 + i0) * L2 + j);
  const float* rM = rowM + (size_t)b * L1;
  const float* rS = rowS + (size_t)b * L1;

#if HAVE_ASYNC_LDS
  __shared__ v4f stage[2][256];
  async_copy16(&stage[0][tid], srow);
  int buf = 0;
#endif

  #pragma unroll
  for (int r = 0; r < RI; ++r) {
    const int i = i0 + r;
    v4f v;
#if HAVE_ASYNC_LDS
    if (r + 1 < RI) {
      async_copy16(&stage[buf ^ 1][tid], srow + L2);
      __builtin_amdgcn_s_wait_asynccnt(1);
    } else {
      __builtin_amdgcn_s_wait_asynccnt(0);
    }
    asm volatile("" ::: "memory");
    v = stage[buf][tid];        // lane reads back exactly its own 16 B
    buf ^= 1;
#else
    v = __builtin_nontemporal_load((const v4f*)srow);   // last-use read
#endif
    v4f o = z;
    if (i < l0) {                      // uniform branch
      const float rm  = rM[i];
      const float rrs = 1.0f / rS[i];
      const float a0 = rm + cm.x, a1 = rm + cm.y, a2 = rm + cm.z, a3 = rm + cm.w;
      // row_sm*col_sm = exp(2x - rm - cm)/(rs*cs), x = sim * (1/TAU)
      float e0 = __expf(fmaf(v.x, 2.0f * TAU_INV, -a0)) * (rrs * rc0);
      float e1 = __expf(fmaf(v.y, 2.0f * TAU_INV, -a1)) * (rrs * rc1);
      float e2 = __expf(fmaf(v.z, 2.0f * TAU_INV, -a2)) * (rrs * rc2);
      float e3 = __expf(fmaf(v.w, 2.0f * TAU_INV, -a3)) * (rrs * rc3);
      o.x = vj0 ? sqrtf(EPS_V + e0) : 0.f;
      o.y = vj1 ? sqrtf(EPS_V + e1) : 0.f;
      o.z = vj2 ? sqrtf(EPS_V + e2) : 0.f;
      o.w = vj3 ? sqrtf(EPS_V + e3) : 0.f;
    }
    __builtin_nontemporal_store(o, (v4f*)orow);
    srow += L2; orow += L2;
  }
}

extern "C" void kernel_launch(void* const* d_in, const int* in_sizes, int n_in,
                              void* d_out, int out_size, void* d_ws, size_t ws_size,
                              hipStream_t stream)
{
  const float* sim     = (const float*)d_in[0];
  const int*   lengths = (const int*)d_in[1];
  float*       out     = (float*)d_out;

  const int Bn    = in_sizes[1] / 2;     // 8
  const int nrows = Bn * L1;             // 16384
  const int ncols = Bn * L2;             // 16384

  float* ws   = (float*)d_ws;
  float* rowM = ws;
  float* rowS = rowM + nrows;
  float* colM = rowS + nrows;
  float* colS = colM + ncols;
  float* partM = colS + ncols;

  // Size the i-split of the column pass to available workspace (cap 32).
  const size_t fixedFloats = (size_t)2 * nrows + (size_t)2 * ncols;
  const size_t totalFloats = ws_size / sizeof(float);
  size_t avail = (totalFloats > fixedFloats) ? (totalFloats - fixedFloats) : 0;
  int split = (int)(avail / ((size_t)2 * ncols));
  if (split > 32) split = 32;
  if (split < 1)  split = 1;
  float* partS = partM + (size_t)split * ncols;
  const int rps = (L1 + split - 1) / split;

  // 1) per-row stats (wave per row)
  row_stats_kernel<<<nrows / 8, 256, 0, stream>>>(sim, lengths, rowM, rowS);

  // 2) per-column partial stats
  dim3 gcol(ncols / 1024, split);
  col_stats_partial_kernel<<<gcol, 256, 0, stream>>>(sim, lengths, partM, partS,
                                                     rps, ncols);

  // 3) combine partials
  col_combine_kernel<<<(ncols + 255) / 256, 256, 0, stream>>>(partM, partS,
                                                              colM, colS,
                                                              split, ncols);

  // 4) fused bidirectional softmax output
  const int nblocksO = Bn * (L1 / 16) * (L2 / 1024);
  bidir_out_kernel<<<nblocksO, 256, 0, stream>>>(sim, lengths, rowM, rowS,
                                                 colM, colS, out);
}